// GIN_52991306498534
// MI455X (gfx1250) — compile-verified
//
#include <hip/hip_runtime.h>

#define N_NODES   100000
#define N_EDGES   1600000
#define N_GRAPHS  64
#define NUM_ATOMS 64
#define HID       128
#define POOLW     (3 * HID)
#define NUM_CLASSES 7
#define BN_EPS    1e-5f

typedef float v2f __attribute__((ext_vector_type(2)));
typedef float v8f __attribute__((ext_vector_type(8)));

// ---------------------------------------------------------------------------
// WMMA k-loop: one wave computes a 16(M) x 64(N) strip as 4 accumulators,
// A fragments from LDS (stride AS, AS%64==4 -> conflict-free ds_load b64),
// B fragments from global weight matrix W[K][ldw] (L2-resident, 64KB).
// Layout per ISA 7.12.2 (f32 16x16x4):
//   A: lanes 0-15 row M=l16 hold K=k0,k0+1 ; lanes 16-31 hold K=k0+2,k0+3
//   B: lanes 0-15 col N=l16 hold K=k0,k0+1 ; lanes 16-31 hold K=k0+2,k0+3
//   C/D: vgpr r -> M = r + 8*half, N = l16
// ---------------------------------------------------------------------------
__device__ __forceinline__ void wmma_kloop(const float* zs, int AS, int K,
                                           const float* __restrict__ W, int ldw,
                                           int m_base, int n_base,
                                           int l16, int half, v8f acc[4])
{
    const int arow = (m_base + l16) * AS;
    for (int k0 = 0; k0 < K; k0 += 4) {
        v2f a;
        const int ai = arow + k0 + 2 * half;
        a.x = zs[ai];
        a.y = zs[ai + 1];
        const int kb = k0 + 2 * half;
#pragma unroll
        for (int j = 0; j < 4; ++j) {
            const int col = n_base + j * 16 + l16;
            v2f b;
            b.x = W[(size_t)kb * ldw + col];
            b.y = W[(size_t)(kb + 1) * ldw + col];
            acc[j] = __builtin_amdgcn_wmma_f32_16x16x4_f32(
                false, a, false, b, (short)0, acc[j], false, false);
        }
    }
}

// ---------------------------------------------------------------------------
// Embedding: h0 = x @ emb_w + emb_b   ([N,64] x [64,128])
// 64-node tile per 256-thread block (8 waves; waves 0-3 M-strips at n_base 0,
// waves 4-7 same M-strips at n_base 64). Uniform full-tile fast path: only
// the last block (partial tile) pays per-element guards.
// ---------------------------------------------------------------------------
__global__ __launch_bounds__(256) void emb_kernel(const float* __restrict__ x,
                                                  const float* __restrict__ w,
                                                  const float* __restrict__ bias,
                                                  float* __restrict__ hout, int nN)
{
    constexpr int MT = 64, K = NUM_ATOMS, AS = 68; // 68 % 64 == 4
    __shared__ float zs[MT * AS];
    const int tid = threadIdx.x, blk = blockIdx.x;
    const bool full = (blk * MT + MT) <= nN; // uniform per block

    if (full) {
#pragma unroll
        for (int i = tid; i < MT * (K / 4); i += 256) {
            const int r = i >> 4, c4 = i & 15;
            const float4 v = ((const float4*)(x + (size_t)(blk * MT + r) * K))[c4];
            float* zp = &zs[r * AS + c4 * 4];
            zp[0] = v.x; zp[1] = v.y; zp[2] = v.z; zp[3] = v.w;
        }
    } else {
        for (int i = tid; i < MT * (K / 4); i += 256) {
            const int r = i >> 4, c4 = i & 15;
            const int row = blk * MT + r;
            float4 v = make_float4(0.f, 0.f, 0.f, 0.f);
            if (row < nN) v = ((const float4*)(x + (size_t)row * K))[c4];
            float* zp = &zs[r * AS + c4 * 4];
            zp[0] = v.x; zp[1] = v.y; zp[2] = v.z; zp[3] = v.w;
        }
    }
    __syncthreads();

    const int wave = tid >> 5, lane = tid & 31, half = lane >> 4, l16 = lane & 15;
    const int m_base = (wave & 3) * 16;
    const int n_base = (wave >> 2) * 64;

    v8f acc[4];
#pragma unroll
    for (int j = 0; j < 4; ++j)
#pragma unroll
        for (int r = 0; r < 8; ++r) acc[j][r] = 0.f;

    wmma_kloop(zs, AS, K, w, HID, m_base, n_base, l16, half, acc);

    if (full) {
#pragma unroll
        for (int j = 0; j < 4; ++j) {
            const int col = n_base + j * 16 + l16;
            const float b = bias[col];
#pragma unroll
            for (int r = 0; r < 8; ++r) {
                const size_t row = (size_t)(blk * MT + m_base + r + 8 * half);
                hout[row * HID + col] = acc[j][r] + b;
            }
        }
    } else {
#pragma unroll
        for (int j = 0; j < 4; ++j) {
            const int col = n_base + j * 16 + l16;
            const float b = bias[col];
#pragma unroll
            for (int r = 0; r < 8; ++r) {
                const int row = blk * MT + m_base + r + 8 * half;
                if (row < nN) hout[(size_t)row * HID + col] = acc[j][r] + b;
            }
        }
    }
}

// ---------------------------------------------------------------------------
// Fused GIN MLP layer:
//   z = h + agg (staged to LDS) ; t = relu(BN(z@w1+b1)) (written back to LDS,
//   barrier-separated) ; out = relu(t@w2+b2)
// ---------------------------------------------------------------------------
__global__ __launch_bounds__(256) void gin_mlp_kernel(
    const float* __restrict__ h, const float* __restrict__ agg,
    const float* __restrict__ w1, const float* __restrict__ b1,
    const float* __restrict__ bg, const float* __restrict__ bb,
    const float* __restrict__ bm, const float* __restrict__ bv,
    const float* __restrict__ w2, const float* __restrict__ b2,
    float* __restrict__ hout, int nN)
{
    constexpr int MT = 64, AS = 132; // 132 % 64 == 4 -> conflict-free b64 A reads
    __shared__ float zs[MT * AS];
    const int tid = threadIdx.x, blk = blockIdx.x;
    const bool full = (blk * MT + MT) <= nN; // uniform per block

    if (full) {
#pragma unroll
        for (int i = tid; i < MT * (HID / 4); i += 256) {
            const int r = i >> 5, c4 = i & 31;
            const size_t row = (size_t)(blk * MT + r);
            const float4 a = ((const float4*)(h + row * HID))[c4];
            const float4 g = ((const float4*)(agg + row * HID))[c4];
            float* zp = &zs[r * AS + c4 * 4];
            zp[0] = a.x + g.x; zp[1] = a.y + g.y;
            zp[2] = a.z + g.z; zp[3] = a.w + g.w;
        }
    } else {
        for (int i = tid; i < MT * (HID / 4); i += 256) {
            const int r = i >> 5, c4 = i & 31;
            const int row = blk * MT + r;
            float4 v = make_float4(0.f, 0.f, 0.f, 0.f);
            if (row < nN) {
                const float4 a = ((const float4*)(h + (size_t)row * HID))[c4];
                const float4 g = ((const float4*)(agg + (size_t)row * HID))[c4];
                v = make_float4(a.x + g.x, a.y + g.y, a.z + g.z, a.w + g.w);
            }
            float* zp = &zs[r * AS + c4 * 4];
            zp[0] = v.x; zp[1] = v.y; zp[2] = v.z; zp[3] = v.w;
        }
    }
    __syncthreads();

    const int wave = tid >> 5, lane = tid & 31, half = lane >> 4, l16 = lane & 15;
    const int m_base = (wave & 3) * 16;
    const int n_base = (wave >> 2) * 64;

    v8f acc[4];
#pragma unroll
    for (int j = 0; j < 4; ++j)
#pragma unroll
        for (int r = 0; r < 8; ++r) acc[j][r] = 0.f;

    wmma_kloop(zs, AS, HID, w1, HID, m_base, n_base, l16, half, acc);

    __syncthreads(); // all GEMM1 LDS reads complete before overwriting zs with t

    // folded BatchNorm + ReLU; write t tile back into zs
#pragma unroll
    for (int j = 0; j < 4; ++j) {
        const int col = n_base + j * 16 + l16;
        const float s = bg[col] * rsqrtf(bv[col] + BN_EPS);
        const float sh = bb[col] + (b1[col] - bm[col]) * s;
#pragma unroll
        for (int r = 0; r < 8; ++r) {
            float z = acc[j][r] * s + sh;
            z = fmaxf(z, 0.f);
            zs[(m_base + r + 8 * half) * AS + col] = z;
            acc[j][r] = 0.f;
        }
    }
    __syncthreads();

    wmma_kloop(zs, AS, HID, w2, HID, m_base, n_base, l16, half, acc);

    if (full) {
#pragma unroll
        for (int j = 0; j < 4; ++j) {
            const int col = n_base + j * 16 + l16;
            const float b = b2[col];
#pragma unroll
            for (int r = 0; r < 8; ++r) {
                const size_t row = (size_t)(blk * MT + m_base + r + 8 * half);
                hout[row * HID + col] = fmaxf(acc[j][r] + b, 0.f);
            }
        }
    } else {
#pragma unroll
        for (int j = 0; j < 4; ++j) {
            const int col = n_base + j * 16 + l16;
            const float b = b2[col];
#pragma unroll
            for (int r = 0; r < 8; ++r) {
                const int row = blk * MT + m_base + r + 8 * half;
                if (row < nN) hout[(size_t)row * HID + col] = fmaxf(acc[j][r] + b, 0.f);
            }
        }
    }
}

// ---------------------------------------------------------------------------
// Edge aggregation: agg[dst] += h[src]  (32 lanes x float4 per edge)
// ---------------------------------------------------------------------------
__global__ __launch_bounds__(256) void scatter_kernel(
    const float* __restrict__ h, const long long* __restrict__ src,
    const long long* __restrict__ dst, float* __restrict__ agg)
{
    const long long gid = (long long)blockIdx.x * 256 + threadIdx.x;
    const long long e = gid >> 5;
    const int lane = (int)(gid & 31);
    if (e >= N_EDGES) return;
    const long long s = src[e], d = dst[e];
    const float4 v = ((const float4*)(h + s * HID))[lane];
    float* a = agg + d * HID + (size_t)lane * 4;
    atomicAdd(a + 0, v.x);
    atomicAdd(a + 1, v.y);
    atomicAdd(a + 2, v.z);
    atomicAdd(a + 3, v.w);
}

// ---------------------------------------------------------------------------
// Per-layer segment-sum pooling into graph_emb columns [off, off+128)
// ---------------------------------------------------------------------------
__global__ __launch_bounds__(256) void pool_kernel(const float* __restrict__ h,
                                                   const long long* __restrict__ batch,
                                                   float* __restrict__ pooled, int off)
{
    const long long gid = (long long)blockIdx.x * 256 + threadIdx.x;
    const long long n = gid >> 5;
    const int lane = (int)(gid & 31);
    if (n >= N_NODES) return;
    const int g = (int)batch[n];
    const float4 v = ((const float4*)(h + n * HID))[lane];
    float* p = pooled + (size_t)g * POOLW + off + (size_t)lane * 4;
    atomicAdd(p + 0, v.x);
    atomicAdd(p + 1, v.y);
    atomicAdd(p + 2, v.z);
    atomicAdd(p + 3, v.w);
}

__global__ __launch_bounds__(256) void count_kernel(const long long* __restrict__ batch,
                                                    float* __restrict__ counts)
{
    const int n = blockIdx.x * 256 + threadIdx.x;
    if (n < N_NODES) atomicAdd(&counts[(int)batch[n]], 1.f);
}

__global__ __launch_bounds__(256) void div_kernel(float* __restrict__ pooled,
                                                  const float* __restrict__ counts)
{
    const int i = blockIdx.x * 256 + threadIdx.x;
    if (i >= N_GRAPHS * POOLW) return;
    const float d = fmaxf(counts[i / POOLW], 1.f);
    pooled[i] /= d;
}

// ---------------------------------------------------------------------------
// Heads (tiny: 64 graphs) — scalar dot products are fine
// ---------------------------------------------------------------------------
__global__ __launch_bounds__(256) void head1_kernel(const float* __restrict__ ge,
                                                    const float* __restrict__ w,
                                                    const float* __restrict__ b,
                                                    float* __restrict__ hgm)
{
    const int i = blockIdx.x * 256 + threadIdx.x;
    if (i >= N_GRAPHS * POOLW) return;
    const int g = i / POOLW, c = i % POOLW;
    float s = b[c];
    const float* row = ge + (size_t)g * POOLW;
    for (int k = 0; k < POOLW; ++k) s += row[k] * w[(size_t)k * POOLW + c];
    hgm[i] = fmaxf(s, 0.f);
}

__global__ __launch_bounds__(256) void head2_kernel(const float* __restrict__ hgm,
                                                    const float* __restrict__ w,
                                                    const float* __restrict__ b,
                                                    float* __restrict__ out)
{
    const int i = blockIdx.x * 256 + threadIdx.x;
    if (i >= N_GRAPHS * NUM_CLASSES) return;
    const int g = i / NUM_CLASSES, c = i % NUM_CLASSES;
    float s = b[c];
    const float* row = hgm + (size_t)g * POOLW;
    for (int k = 0; k < POOLW; ++k) s += row[k] * w[(size_t)k * NUM_CLASSES + c];
    out[i] = s;
}

// ---------------------------------------------------------------------------
extern "C" void kernel_launch(void* const* d_in, const int* in_sizes, int n_in,
                              void* d_out, int out_size, void* d_ws, size_t ws_size,
                              hipStream_t stream)
{
    const float* x        = (const float*)d_in[0];
    const long long* ei   = (const long long*)d_in[2];
    const long long* batch= (const long long*)d_in[3];
    const float* emb_w    = (const float*)d_in[4];
    const float* emb_b    = (const float*)d_in[5];
    const float* conv_w1  = (const float*)d_in[6];
    const float* conv_b1  = (const float*)d_in[7];
    const float* bn_g     = (const float*)d_in[8];
    const float* bn_b     = (const float*)d_in[9];
    const float* bn_m     = (const float*)d_in[10];
    const float* bn_v     = (const float*)d_in[11];
    const float* conv_w2  = (const float*)d_in[12];
    const float* conv_b2  = (const float*)d_in[13];
    const float* lin1_w   = (const float*)d_in[14];
    const float* lin1_b   = (const float*)d_in[15];
    const float* lin2_w   = (const float*)d_in[16];
    const float* lin2_b   = (const float*)d_in[17];

    float* ws = (float*)d_ws;
    const size_t NH = (size_t)N_NODES * HID;
    float* hA     = ws;
    float* hB     = ws + NH;
    float* agg    = ws + 2 * NH;
    float* counts = ws + 3 * NH;
    float* hgm    = counts + N_GRAPHS;

    float* ge   = (float*)d_out;                 // graph_emb [64,384]
    float* out2 = ge + (size_t)N_GRAPHS * POOLW; // logits [64,7]

    hipMemsetAsync(d_out, 0, sizeof(float) * (size_t)out_size, stream);
    hipMemsetAsync(counts, 0, sizeof(float) * N_GRAPHS, stream);

    const int nbG = (N_NODES + 63) / 64;
    emb_kernel<<<nbG, 256, 0, stream>>>(x, emb_w, emb_b, hA, N_NODES);

    float* cur = hA;
    float* nxt = hB;
    for (int i = 0; i < 3; ++i) {
        hipMemsetAsync(agg, 0, sizeof(float) * NH, stream);
        scatter_kernel<<<(N_EDGES * 32) / 256, 256, 0, stream>>>(cur, ei, ei + N_EDGES, agg);
        gin_mlp_kernel<<<nbG, 256, 0, stream>>>(
            cur, agg,
            conv_w1 + (size_t)i * HID * HID, conv_b1 + i * HID,
            bn_g + i * HID, bn_b + i * HID, bn_m + i * HID, bn_v + i * HID,
            conv_w2 + (size_t)i * HID * HID, conv_b2 + i * HID,
            nxt, N_NODES);
        pool_kernel<<<(N_NODES * 32) / 256, 256, 0, stream>>>(nxt, batch, ge, i * HID);
        float* t = cur; cur = nxt; nxt = t;
    }

    count_kernel<<<(N_NODES + 255) / 256, 256, 0, stream>>>(batch, counts);
    div_kernel<<<(N_GRAPHS * POOLW + 255) / 256, 256, 0, stream>>>(ge, counts);
    head1_kernel<<<(N_GRAPHS * POOLW + 255) / 256, 256, 0, stream>>>(ge, lin1_w, lin1_b, hgm);
    head2_kernel<<<(N_GRAPHS * NUM_CLASSES + 255) / 256, 256, 0, stream>>>(hgm, lin2_w, lin2_b, out2);
}